// GNN_67164698575264
// MI455X (gfx1250) — compile-verified
//
#include <hip/hip_runtime.h>
#include <math.h>

typedef __attribute__((ext_vector_type(2))) float v2f;
typedef __attribute__((ext_vector_type(8))) float v8f;

#define Bn 8
#define Nn 4096
#define Cn 128
#define Hn 256
#define Kn 16
#define Mn (Bn * Nn)

// ---------------------------------------------------------------------------
// 1) kNN: one thread per target node, candidates staged through LDS in tiles
//    of 256. Top-16 kept register-resident: replacement + max-rescan are fully
//    unrolled selects (no dynamic indexing -> no scratch).
// ---------------------------------------------------------------------------
__global__ __launch_bounds__(256) void knn_kernel(const float* __restrict__ pos,
                                                  int* __restrict__ knn) {
    __shared__ float sp[256 * 3];
    const int tid = threadIdx.x;
    const int i   = blockIdx.x * 256 + tid;   // global node id, 0..M-1
    const int b   = i / Nn;                   // all threads in block share b
    const int jbase = b * Nn;

    const float px = pos[i * 3 + 0];
    const float py = pos[i * 3 + 1];
    const float pz = pos[i * 3 + 2];

    float bd[Kn];
    int   bi[Kn];
#pragma unroll
    for (int s = 0; s < Kn; ++s) { bd[s] = 3.4e38f; bi[s] = 0; }
    float worst = 3.4e38f;
    int   wslot = 0;

    for (int j0 = 0; j0 < Nn; j0 += 256) {
        sp[tid * 3 + 0] = pos[(jbase + j0 + tid) * 3 + 0];
        sp[tid * 3 + 1] = pos[(jbase + j0 + tid) * 3 + 1];
        sp[tid * 3 + 2] = pos[(jbase + j0 + tid) * 3 + 2];
        __syncthreads();

        for (int jj = 0; jj < 256; ++jj) {
            const float dx = px - sp[jj * 3 + 0];
            const float dy = py - sp[jj * 3 + 1];
            const float dz = pz - sp[jj * 3 + 2];
            const float d2 = dx * dx + dy * dy + dz * dz;
            if (d2 < worst) {
                const int j = j0 + jj;
#pragma unroll
                for (int s = 0; s < Kn; ++s)
                    if (s == wslot) { bd[s] = d2; bi[s] = j; }
                worst = -1.0f;
#pragma unroll
                for (int s = 0; s < Kn; ++s)
                    if (bd[s] > worst) { worst = bd[s]; wslot = s; }
            }
        }
        __syncthreads();
    }

#pragma unroll
    for (int s = 0; s < Kn; ++s) knn[i * Kn + s] = bi[s];
}

// ---------------------------------------------------------------------------
// 2) Fused QKV GEMM: a_src = x@W_src, a_dst = x@W_dst, v = x@W_val.
//    Register-blocked: block = 256 threads = 8 wave32 owns a 64-row stripe of
//    x (staged in LDS, padded stride 132 -> conflict-free fragment reads,
//    16B-aligned float4 fill). Wave w owns output columns [16w, 16w+16) and
//    4 M-tiles: each B fragment (2 global loads) feeds 4 back-to-back
//    v_wmma_f32_16x16x4_f32 ops against 4 A fragments -> 4x fewer weight
//    loads per WMMA than a 16-row block.
// ---------------------------------------------------------------------------
__global__ __launch_bounds__(256) void qkv_kernel(
        const float* __restrict__ x,
        const float* __restrict__ Wsrc, const float* __restrict__ Wdst,
        const float* __restrict__ Wval,
        float* __restrict__ a_src, float* __restrict__ a_dst,
        float* __restrict__ v_out) {
    __shared__ float xs[64 * 132];
    const int tid = threadIdx.x;
    const int m0  = blockIdx.x * 64;

    // cooperative 64x128 tile load, float4 per transaction
    for (int t = tid; t < 64 * (Cn / 4); t += 256) {
        const int r  = t >> 5;             // 32 float4 per row
        const int c4 = (t & 31) * 4;
        *(float4*)&xs[r * 132 + c4] =
            *(const float4*)&x[(size_t)(m0 + r) * Cn + c4];
    }
    __syncthreads();

    const int wave = tid >> 5;
    const int lane = tid & 31;
    const int half = lane >> 4;            // 0: K pair {0,1}, 1: K pair {2,3}
    const int l16  = lane & 15;
    const int colg = wave * 16 + l16;      // output column (0..127)

    const float* Ws[3] = {Wsrc, Wdst, Wval};
    float*       Os[3] = {a_src, a_dst, v_out};

#pragma unroll
    for (int wi = 0; wi < 3; ++wi) {
        const float* __restrict__ W = Ws[wi];
        v8f acc[4] = {};
#pragma unroll
        for (int kk = 0; kk < Cn; kk += 4) {
            const int k0 = kk + half * 2;
            v2f bf;
            bf[0] = W[(size_t)k0 * Cn + colg];
            bf[1] = W[(size_t)(k0 + 1) * Cn + colg];
#pragma unroll
            for (int mt = 0; mt < 4; ++mt) {
                v2f a;
                a[0] = xs[(mt * 16 + l16) * 132 + k0];
                a[1] = xs[(mt * 16 + l16) * 132 + k0 + 1];
                acc[mt] = __builtin_amdgcn_wmma_f32_16x16x4_f32(
                    false, a, false, bf, (short)0, acc[mt], false, false);
            }
        }
        float* __restrict__ O = Os[wi];
#pragma unroll
        for (int mt = 0; mt < 4; ++mt) {
            const int mb = m0 + mt * 16 + half * 8;
#pragma unroll
            for (int r2 = 0; r2 < 8; ++r2)
                O[(size_t)(mb + r2) * Cn + colg] = acc[mt][r2];
        }
    }
}

// ---------------------------------------------------------------------------
// 3) Attention: one wave per node; lane owns 4 channels (32*4 = 128).
//    Two-pass safe softmax over k=16; delta (3 FMAs) recomputed in pass 2
//    instead of stored, keeping register pressure low. All gathers are
//    float4-coalesced (lane*4 channels -> 512B contiguous per wave).
// ---------------------------------------------------------------------------
__global__ __launch_bounds__(256) void attn_kernel(
        const float* __restrict__ pos, const int* __restrict__ knn,
        const float* __restrict__ a_src, const float* __restrict__ a_dst,
        const float* __restrict__ v, const float* __restrict__ W_pos,
        const float* __restrict__ b_pos, float* __restrict__ h) {
    const int tid  = threadIdx.x;
    const int wave = tid >> 5;
    const int lane = tid & 31;
    const int i    = blockIdx.x * 8 + wave;   // node 0..M-1
    const int b    = i / Nn;
    const int c0   = lane * 4;

    const float pix = pos[i * 3 + 0];
    const float piy = pos[i * 3 + 1];
    const float piz = pos[i * 3 + 2];

    float ad[4], wp0[4], wp1[4], wp2[4], bp[4];
    *(float4*)ad  = *(const float4*)(a_dst + (size_t)i * Cn + c0);
    *(float4*)wp0 = *(const float4*)(W_pos + 0 * Cn + c0);
    *(float4*)wp1 = *(const float4*)(W_pos + 1 * Cn + c0);
    *(float4*)wp2 = *(const float4*)(W_pos + 2 * Cn + c0);
    *(float4*)bp  = *(const float4*)(b_pos + c0);

    int jg[Kn];
#pragma unroll
    for (int k = 0; k < Kn; ++k) jg[k] = b * Nn + knn[i * Kn + k];

    // pass 1: per-channel max over neighbors
    float m[4] = {-3.4e38f, -3.4e38f, -3.4e38f, -3.4e38f};
#pragma unroll
    for (int k = 0; k < Kn; ++k) {
        const int g = jg[k];
        const float rx = pix - pos[g * 3 + 0];
        const float ry = piy - pos[g * 3 + 1];
        const float rz = piz - pos[g * 3 + 2];
        float as[4];
        *(float4*)as = *(const float4*)(a_src + (size_t)g * Cn + c0);
#pragma unroll
        for (int t = 0; t < 4; ++t) {
            const float d  = rx * wp0[t] + ry * wp1[t] + rz * wp2[t] + bp[t];
            const float al = ad[t] - as[t] + d;
            m[t] = fmaxf(m[t], al);
        }
    }

    // pass 2: exp-sum and weighted value accumulation
    float s[4] = {0.f, 0.f, 0.f, 0.f};
    float o[4] = {0.f, 0.f, 0.f, 0.f};
#pragma unroll
    for (int k = 0; k < Kn; ++k) {
        const int g = jg[k];
        const float rx = pix - pos[g * 3 + 0];
        const float ry = piy - pos[g * 3 + 1];
        const float rz = piz - pos[g * 3 + 2];
        float as[4], vv[4];
        *(float4*)as = *(const float4*)(a_src + (size_t)g * Cn + c0);
        *(float4*)vv = *(const float4*)(v + (size_t)g * Cn + c0);
#pragma unroll
        for (int t = 0; t < 4; ++t) {
            const float d  = rx * wp0[t] + ry * wp1[t] + rz * wp2[t] + bp[t];
            const float al = ad[t] - as[t] + d;
            const float e  = __expf(al - m[t]);
            s[t] += e;
            o[t] += e * (vv[t] + d);
        }
    }

    float res[4];
#pragma unroll
    for (int t = 0; t < 4; ++t) res[t] = o[t] / s[t];
    *(float4*)(h + (size_t)i * Cn + c0) = *(const float4*)res;
}

// ---------------------------------------------------------------------------
// 4) Output GEMM with bias: y = h @ W_out + b_out.  [32768,128] x [128,256].
//    Same register-blocked WMMA structure: 64-row block, grid.y selects a
//    128-column slab, each B fragment reused across 4 M-tiles.
// ---------------------------------------------------------------------------
__global__ __launch_bounds__(256) void out_gemm_kernel(
        const float* __restrict__ h, const float* __restrict__ W_out,
        const float* __restrict__ b_out, float* __restrict__ y) {
    __shared__ float xs[64 * 132];
    const int tid = threadIdx.x;
    const int m0  = blockIdx.x * 64;
    const int nb  = blockIdx.y * 128;

    for (int t = tid; t < 64 * (Cn / 4); t += 256) {
        const int r  = t >> 5;
        const int c4 = (t & 31) * 4;
        *(float4*)&xs[r * 132 + c4] =
            *(const float4*)&h[(size_t)(m0 + r) * Cn + c4];
    }
    __syncthreads();

    const int wave = tid >> 5;
    const int lane = tid & 31;
    const int half = lane >> 4;
    const int l16  = lane & 15;
    const int colg = nb + wave * 16 + l16;   // output column (0..255)

    v8f acc[4] = {};
#pragma unroll
    for (int kk = 0; kk < Cn; kk += 4) {
        const int k0 = kk + half * 2;
        v2f bf;
        bf[0] = W_out[(size_t)k0 * Hn + colg];
        bf[1] = W_out[(size_t)(k0 + 1) * Hn + colg];
#pragma unroll
        for (int mt = 0; mt < 4; ++mt) {
            v2f a;
            a[0] = xs[(mt * 16 + l16) * 132 + k0];
            a[1] = xs[(mt * 16 + l16) * 132 + k0 + 1];
            acc[mt] = __builtin_amdgcn_wmma_f32_16x16x4_f32(
                false, a, false, bf, (short)0, acc[mt], false, false);
        }
    }

    const float bias = b_out[colg];
#pragma unroll
    for (int mt = 0; mt < 4; ++mt) {
        const int mb = m0 + mt * 16 + half * 8;
#pragma unroll
        for (int r2 = 0; r2 < 8; ++r2)
            y[(size_t)(mb + r2) * Hn + colg] = acc[mt][r2] + bias;
    }
}

// ---------------------------------------------------------------------------
extern "C" void kernel_launch(void* const* d_in, const int* in_sizes, int n_in,
                              void* d_out, int out_size, void* d_ws, size_t ws_size,
                              hipStream_t stream) {
    (void)in_sizes; (void)n_in; (void)out_size; (void)ws_size;

    const float* x     = (const float*)d_in[0];  // [B,N,C]
    const float* pos   = (const float*)d_in[1];  // [B,N,3]
    const float* W_src = (const float*)d_in[2];  // [C,C]
    const float* W_dst = (const float*)d_in[3];  // [C,C]
    const float* W_val = (const float*)d_in[4];  // [C,C]
    const float* W_pos = (const float*)d_in[5];  // [3,C]
    const float* b_pos = (const float*)d_in[6];  // [C]
    const float* W_out = (const float*)d_in[7];  // [C,H]
    const float* b_out = (const float*)d_in[8];  // [H]
    float*       y     = (float*)d_out;          // [B,N,H]

    // workspace: 4 fp32 [M,C] buffers (64 MB) + int [M,K] (2 MB)
    float* a_src = (float*)d_ws;
    float* a_dst = a_src + (size_t)Mn * Cn;
    float* v     = a_dst + (size_t)Mn * Cn;
    float* h     = v     + (size_t)Mn * Cn;
    int*   knn   = (int*)(h + (size_t)Mn * Cn);

    knn_kernel<<<Mn / 256, 256, 0, stream>>>(pos, knn);
    qkv_kernel<<<Mn / 64, 256, 0, stream>>>(x, W_src, W_dst, W_val,
                                            a_src, a_dst, v);
    attn_kernel<<<Mn / 8, 256, 0, stream>>>(pos, knn, a_src, a_dst, v,
                                            W_pos, b_pos, h);
    out_gemm_kernel<<<dim3(Mn / 64, Hn / 128), 256, 0, stream>>>(h, W_out,
                                                                 b_out, y);
}